// EdgeValuePredictor_78761110274681
// MI455X (gfx1250) — compile-verified
//
#include <hip/hip_runtime.h>
#include <hip/hip_bf16.h>

// CDNA5 / gfx1250, wave32 only.
typedef float v2f __attribute__((ext_vector_type(2)));
typedef float v8f __attribute__((ext_vector_type(8)));

#define D_H 128

__device__ __forceinline__ v8f wmma4(v2f a, v2f b, v8f c) {
  return __builtin_amdgcn_wmma_f32_16x16x4_f32(false, a, false, b,
                                               (short)0, c, false, false);
}

// ---------------------------------------------------------------------------
// Pre-pack weight matrix (krows x 128, row-major) into K-pair-interleaved
// layout: Wp[(k/2)*256 + 2n] = W[k][n], Wp[(k/2)*256 + 2n + 1] = W[k+1][n].
// A WMMA B-fragment {W[ka][n], W[ka+1][n]} is then one contiguous 8B word.
// ---------------------------------------------------------------------------
__global__ void pack_weights_kernel(const float* __restrict__ W,
                                    float* __restrict__ Wp,
                                    int krows) {
  int t = blockIdx.x * blockDim.x + threadIdx.x;
  int total = (krows >> 1) * D_H;          // kpairs * 128
  if (t >= total) return;
  int kp = t >> 7;                         // k-pair index
  int n  = t & 127;
  float2 v;
  v.x = W[(2 * kp) * D_H + n];
  v.y = W[(2 * kp + 1) * D_H + n];
  *(float2*)(Wp + 2 * t) = v;
}

// ---------------------------------------------------------------------------
// Stage a packed 64KB weight panel global -> LDS with async-LDS loads
// (ASYNCcnt-tracked). 256 threads x 16 b128 ops. Call uniformly per block.
// ---------------------------------------------------------------------------
__device__ __forceinline__ void stage_panel_async(const float* __restrict__ Wp,
                                                  float* sW) {
  __syncthreads();                         // previous readers done with LDS
  const unsigned t = threadIdx.x;          // 0..255
#pragma unroll
  for (int i = 0; i < 16; ++i) {
    const unsigned fidx = (i * 256u + t) * 4u;            // float index
    const unsigned lds_off = (unsigned)(uintptr_t)(sW + fidx);
    const float* g = Wp + fidx;
    asm volatile("global_load_async_to_lds_b128 %0, %1, off"
                 :: "v"(lds_off), "v"(g) : "memory");
  }
  asm volatile("s_wait_asynccnt 0x0" ::: "memory");
  __syncthreads();                         // all waves' copies visible
}

// ---------------------------------------------------------------------------
// Accumulate 16x128 += Arow(16x128) @ W(128x128), fp32 WMMA 16x16x4.
// B-fragments: single ds_load_b64 from the packed LDS panel.
// ---------------------------------------------------------------------------
__device__ __forceinline__ void gemm_k128(const float* __restrict__ arow,
                                          const float* sW,   // packed panel
                                          int half, int l16, v8f acc[8]) {
  for (int k0 = 0; k0 < D_H; k0 += 4) {
    const int ka = k0 + 2 * half;
    v2f a; a.x = arow[ka]; a.y = arow[ka + 1];
    // packed index for (kpair = k0/2 + half, n = j*16 + l16):
    const float* bbase = sW + ((k0 >> 1) + half) * (2 * D_H) + 2 * l16;
#pragma unroll
    for (int j = 0; j < 8; ++j) {
      v2f b = *(const v2f*)(bbase + j * 32);
      acc[j] = wmma4(a, b, acc[j]);
    }
  }
}

// ---------------------------------------------------------------------------
// Scatter-add: agg[dst[e]][:] += x[src[e]][:]   (one wave per edge)
// ---------------------------------------------------------------------------
__global__ void scatter_add_kernel(const float* __restrict__ xin,
                                   const long long* __restrict__ src,
                                   const long long* __restrict__ dst,
                                   float* __restrict__ agg,
                                   int nedges) {
  long long t = (long long)blockIdx.x * blockDim.x + threadIdx.x;
  int e = (int)(t >> 5);
  if (e >= nedges) return;
  int q = (int)(t & 31) * 4;
  long long s = src[e];
  long long d = dst[e];
  const float4 v = *(const float4*)(xin + s * D_H + q);
  float* p = agg + d * D_H + q;
  atomicAdd(p + 0, v.x);
  atomicAdd(p + 1, v.y);
  atomicAdd(p + 2, v.z);
  atomicAdd(p + 3, v.w);
}

// ---------------------------------------------------------------------------
// GraphConv: out = relu(agg @ Wrel + bias + xin @ Wroot)
// Phases: stage Wrel(packed) -> accumulate; stage Wroot -> accumulate.
// ---------------------------------------------------------------------------
__global__ void node_conv_kernel(const float* __restrict__ agg,
                                 const float* __restrict__ xin,
                                 const float* __restrict__ Wrelp,
                                 const float* __restrict__ Wrootp,
                                 const float* __restrict__ bias,
                                 float* __restrict__ out,
                                 int nrows) {
  __shared__ float sW[D_H * D_H];          // 64 KB
  const int wave = threadIdx.x >> 5;
  const int lane = threadIdx.x & 31;
  const int ntiles = nrows >> 4;
  int tile = blockIdx.x * (blockDim.x >> 5) + wave;
  if (tile >= ntiles) tile = ntiles - 1;   // clamp (no barrier divergence)
  const int row0 = tile * 16;
  const int half = lane >> 4;
  const int l16  = lane & 15;

  const float* arow = agg + (long long)(row0 + l16) * D_H;
  const float* xrow = xin + (long long)(row0 + l16) * D_H;

  v8f acc[8] = {};
  stage_panel_async(Wrelp, sW);
  gemm_k128(arow, sW, half, l16, acc);
  stage_panel_async(Wrootp, sW);
  gemm_k128(xrow, sW, half, l16, acc);

#pragma unroll
  for (int j = 0; j < 8; ++j) {
    float bv = bias[j * 16 + l16];
#pragma unroll
    for (int r = 0; r < 8; ++r) {
      float v = acc[j][r] + bv;
      v = v > 0.f ? v : 0.f;
      out[(long long)(row0 + r + 8 * half) * D_H + j * 16 + l16] = v;
    }
  }
}

// ---------------------------------------------------------------------------
// Head epilogue: relu(acc + b1) then 128->1 dot with W2, 16-lane reduction.
// ---------------------------------------------------------------------------
__device__ __forceinline__ void head_epilogue(const v8f acc[8],
                                              const float* __restrict__ b1v,
                                              const float* __restrict__ W2,
                                              float b2s, int l16,
                                              float out8[8]) {
  float dot[8];
#pragma unroll
  for (int r = 0; r < 8; ++r) dot[r] = 0.f;
#pragma unroll
  for (int j = 0; j < 8; ++j) {
    float bv = b1v[j * 16 + l16];
    float w2 = W2[j * 16 + l16];
#pragma unroll
    for (int r = 0; r < 8; ++r) {
      float v = acc[j][r] + bv;
      v = v > 0.f ? v : 0.f;
      dot[r] = fmaf(v, w2, dot[r]);
    }
  }
#pragma unroll
  for (int m = 1; m < 16; m <<= 1) {
#pragma unroll
    for (int r = 0; r < 8; ++r) dot[r] += __shfl_xor(dot[r], m, 32);
  }
#pragma unroll
  for (int r = 0; r < 8; ++r) out8[r] = dot[r] + b2s;
}

// ---------------------------------------------------------------------------
// Edge heads. Phases per block:
//   stage Wm1p[src-half] -> accum(ps); stage Wm1p[dst-half] -> accum(pd) -> mean
//   stage Wv1p[src-half] -> accum(ps); stage Wv1p[dst-half] -> accum(pd) -> var
// ---------------------------------------------------------------------------
__global__ void edge_head_kernel(const float* __restrict__ h,
                                 const long long* __restrict__ src,
                                 const long long* __restrict__ dst,
                                 const float* __restrict__ Wm1p, const float* __restrict__ bm1,
                                 const float* __restrict__ Wm2,  const float* __restrict__ bm2,
                                 const float* __restrict__ Wv1p, const float* __restrict__ bv1,
                                 const float* __restrict__ Wv2,  const float* __restrict__ bv2,
                                 float* __restrict__ mean_out,
                                 float* __restrict__ var_out,
                                 int nedges) {
  __shared__ float sW[D_H * D_H];          // 64 KB
  const int wave = threadIdx.x >> 5;
  const int lane = threadIdx.x & 31;
  const int ntiles = nedges >> 4;
  int tile = blockIdx.x * (blockDim.x >> 5) + wave;
  if (tile >= ntiles) tile = ntiles - 1;   // clamp (no barrier divergence)
  const int e0 = tile * 16;
  const int half = lane >> 4;
  const int l16  = lane & 15;

  const long long s = src[e0 + l16];
  const long long d = dst[e0 + l16];
  const float* ps = h + s * D_H;
  const float* pd = h + d * D_H;

  float m8[8], v8[8];

  {
    v8f acc[8] = {};
    stage_panel_async(Wm1p, sW);                   // k-pairs 0..63 (src part)
    gemm_k128(ps, sW, half, l16, acc);
    stage_panel_async(Wm1p + D_H * D_H, sW);       // k-pairs 64..127 (dst part)
    gemm_k128(pd, sW, half, l16, acc);
    head_epilogue(acc, bm1, Wm2, bm2[0], l16, m8);
  }
  {
    v8f acc[8] = {};
    stage_panel_async(Wv1p, sW);
    gemm_k128(ps, sW, half, l16, acc);
    stage_panel_async(Wv1p + D_H * D_H, sW);
    gemm_k128(pd, sW, half, l16, acc);
    head_epilogue(acc, bv1, Wv2, bv2[0], l16, v8);
  }

  if (l16 == 0) {
#pragma unroll
    for (int r = 0; r < 8; ++r) {
      int e = e0 + r + 8 * half;
      mean_out[e] = m8[r];
      var_out[e]  = expf(0.5f * v8[r]);
    }
  }
}

// ---------------------------------------------------------------------------
extern "C" void kernel_launch(void* const* d_in, const int* in_sizes, int n_in,
                              void* d_out, int out_size, void* d_ws, size_t ws_size,
                              hipStream_t stream) {
  const float*     x    = (const float*)d_in[0];
  const long long* ei   = (const long long*)d_in[1];     // int64 (2, E)
  const float*     W1r  = (const float*)d_in[2];
  const float*     b1   = (const float*)d_in[3];
  const float*     W1o  = (const float*)d_in[4];
  const float*     W2r  = (const float*)d_in[5];
  const float*     b2   = (const float*)d_in[6];
  const float*     W2o  = (const float*)d_in[7];
  const float*     Wm1  = (const float*)d_in[8];
  const float*     bm1  = (const float*)d_in[9];
  const float*     Wm2  = (const float*)d_in[10];
  const float*     bm2  = (const float*)d_in[11];
  const float*     Wv1  = (const float*)d_in[12];
  const float*     bv1  = (const float*)d_in[13];
  const float*     Wv2  = (const float*)d_in[14];
  const float*     bv2  = (const float*)d_in[15];

  const int N = in_sizes[0] / D_H;    // 50000
  const int E = in_sizes[1] / 2;      // 800000
  const long long* src = ei;
  const long long* dst = ei + E;

  float* agg = (float*)d_ws;
  float* h1  = agg + (size_t)N * D_H;
  float* h2  = h1  + (size_t)N * D_H;
  // packed weights after the node buffers (total 128K floats = 512 KB)
  float* W1rp = h2 + (size_t)N * D_H;
  float* W1op = W1rp + D_H * D_H;
  float* W2rp = W1op + D_H * D_H;
  float* W2op = W2rp + D_H * D_H;
  float* Wm1p = W2op + D_H * D_H;
  float* Wv1p = Wm1p + 2 * D_H * D_H;
  const size_t nodeBytes = (size_t)N * D_H * sizeof(float);

  const int scatterBlocks = (int)(((long long)E * 32 + 255) / 256);
  const int convBlocks = ((N / 16) + 7) / 8;           // 8 waves / block
  const int edgeBlocks = ((E / 16) + 7) / 8;

  float* mean_out = (float*)d_out;
  float* var_out  = mean_out + E;

  // Pre-pack weights into K-pair-interleaved layout (tiny, once per launch).
  pack_weights_kernel<<<(64 * D_H + 255) / 256, 256, 0, stream>>>(W1r, W1rp, D_H);
  pack_weights_kernel<<<(64 * D_H + 255) / 256, 256, 0, stream>>>(W1o, W1op, D_H);
  pack_weights_kernel<<<(64 * D_H + 255) / 256, 256, 0, stream>>>(W2r, W2rp, D_H);
  pack_weights_kernel<<<(64 * D_H + 255) / 256, 256, 0, stream>>>(W2o, W2op, D_H);
  pack_weights_kernel<<<(128 * D_H + 255) / 256, 256, 0, stream>>>(Wm1, Wm1p, 2 * D_H);
  pack_weights_kernel<<<(128 * D_H + 255) / 256, 256, 0, stream>>>(Wv1, Wv1p, 2 * D_H);

  // Layer 1
  hipMemsetAsync(agg, 0, nodeBytes, stream);
  scatter_add_kernel<<<scatterBlocks, 256, 0, stream>>>(x, src, dst, agg, E);
  node_conv_kernel<<<convBlocks, 256, 0, stream>>>(agg, x, W1rp, W1op, b1, h1, N);

  // Layer 2
  hipMemsetAsync(agg, 0, nodeBytes, stream);
  scatter_add_kernel<<<scatterBlocks, 256, 0, stream>>>(h1, src, dst, agg, E);
  node_conv_kernel<<<convBlocks, 256, 0, stream>>>(agg, h1, W2rp, W2op, b2, h2, N);

  // Edge heads (mean, var)
  edge_head_kernel<<<edgeBlocks, 256, 0, stream>>>(h2, src, dst,
                                                   Wm1p, bm1, Wm2, bm2,
                                                   Wv1p, bv1, Wv2, bv2,
                                                   mean_out, var_out, E);
}